// MambaModel_6657199309149
// MI455X (gfx1250) — compile-verified
//
#include <hip/hip_runtime.h>

// ---------------------------------------------------------------------------
// Mamba block forward, fused single kernel for gfx1250 (MI455X).
//   grid  = B_SZ blocks (one sequence per workgroup)
//   block = 128 threads = 4 wave32 = (d_inner=8) x (d_state=16) scan lattice
// Dense projections use V_WMMA_F32_16X16X4_F32 (full fp32 precision).
// x staging uses GLOBAL_LOAD_ASYNC_TO_LDS_B128 (double buffered) when the
// toolchain exposes the builtin; falls back to vmem+ds otherwise.
// ---------------------------------------------------------------------------

typedef __attribute__((ext_vector_type(2))) float v2f;
typedef __attribute__((ext_vector_type(8))) float v8f;
typedef __attribute__((ext_vector_type(4))) int   v4i;

// address-space-qualified pointer types for the async global->LDS builtin
typedef __attribute__((address_space(1))) v4i  g_v4i;   // global
typedef __attribute__((address_space(3))) v4i  l_v4i;   // LDS

#define B_SZ     256
#define SEQ_L    4096
#define CHUNK    64
#define NCHUNK   (SEQ_L / CHUNK)
#define NTHREADS 128

// LDS row strides (floats); s_x is unpadded (16B rows) for async B128 writes
#define XS   4    // x tile rows (4 cols, 16B aligned rows for B128)
#define XZS  17   // xz rows (16 cols + pad)
#define US   9    // u rows (8 cols + pad)
#define XDS  49   // x_dbl rows (48 cols + pad; cols 33..47 are zero padding)

#if __has_builtin(__builtin_amdgcn_global_load_async_to_lds_b128)
#define HAS_ASYNC_LDS 1
#endif

__device__ __forceinline__ float fast_sigmoid(float v) {
    // v_exp_f32 + v_add + v_rcp_f32 (no IEEE divide chain)
    return __builtin_amdgcn_rcpf(1.0f + __expf(-v));
}

__device__ __forceinline__ float fast_softplus(float v) {
    return (v > 20.0f) ? v : __logf(1.0f + __expf(v));
}

// Stage one 64-token x tile (64 x 16B) into LDS.
__device__ __forceinline__ void load_x_chunk(const float* gsrc, float* dst, int tid) {
#ifdef HAS_ASYNC_LDS
    for (int i = tid; i < CHUNK; i += NTHREADS) {
        __builtin_amdgcn_global_load_async_to_lds_b128(
            (g_v4i*)(gsrc + i * 4),
            (l_v4i*)(dst  + i * 4),
            0, 0);
    }
#else
    const float4* xv = reinterpret_cast<const float4*>(gsrc);
    for (int i = tid; i < CHUNK; i += NTHREADS) {
        float4 v = xv[i];
        float* r = dst + i * 4;
        r[0] = v.x; r[1] = v.y; r[2] = v.z; r[3] = v.w;
    }
#endif
}

__device__ __forceinline__ void wait_async_x() {
#ifdef HAS_ASYNC_LDS
#if __has_builtin(__builtin_amdgcn_s_wait_asynccnt)
    __builtin_amdgcn_s_wait_asynccnt(0);
#else
    asm volatile("s_wait_asynccnt 0x0" ::: "memory");
#endif
#endif
}

__global__ __launch_bounds__(NTHREADS, 2)
void mamba_fused_kernel(const float* __restrict__ x,        // (B, L, 4)
                        const float* __restrict__ W_in,     // (16, 4)
                        const float* __restrict__ conv_w,   // (8, 1, 4)
                        const float* __restrict__ conv_b,   // (8)
                        const float* __restrict__ W_xproj,  // (33, 8)
                        const float* __restrict__ W_dt,     // (8, 1)
                        const float* __restrict__ b_dt,     // (8)
                        const float* __restrict__ A_log,    // (8, 16)
                        const float* __restrict__ Dp,       // (8)
                        const float* __restrict__ W_out,    // (4, 8)
                        const float* __restrict__ fc_w,     // (3, 4)
                        const float* __restrict__ fc_b,     // (3)
                        float* __restrict__ out)            // (B, L, 3)
{
    __shared__ float s_x [2][CHUNK * XS];   // double-buffered x staging
    __shared__ float s_xz[CHUNK * XZS];
    __shared__ float s_u [CHUNK * US];
    __shared__ float s_xd[CHUNK * XDS];
    __shared__ float s_y [CHUNK * 8];
    __shared__ float s_Win[16 * 4];
    __shared__ float s_Wxp[48 * 8];         // W_xproj padded to 48 rows
    __shared__ float s_cw[8 * 4];
    __shared__ float s_cb[8];
    __shared__ float s_G[3 * 8];            // fc_w @ W_out  (3 x 8)
    __shared__ float s_carry[3 * 8];        // conv state: last 3 tokens' xi

    const int tid  = threadIdx.x;
    const int b    = blockIdx.x;
    const int lane = tid & 31;
    const int wave = tid >> 5;
    const int half = lane >> 4;             // 0: lanes 0-15, 1: lanes 16-31
    const int lidx = lane & 15;
    const int tt0  = wave * 16;             // this wave's 16-token tile

    const float* xb = x + ((size_t)b * SEQ_L) * 4;

    // ---- kick off chunk 0 staging, then stage constants ----------------
    load_x_chunk(xb, s_x[0], tid);

    if (tid < 64) s_Win[tid] = W_in[tid];
    for (int i = tid; i < 48 * 8; i += NTHREADS) {
        int r = i >> 3, c = i & 7;
        s_Wxp[i] = (r < 33) ? W_xproj[r * 8 + c] : 0.0f;
    }
    if (tid < 32) s_cw[tid] = conv_w[tid];
    if (tid < 8)  s_cb[tid] = conv_b[tid];
    if (tid < 24) {
        int o = tid >> 3, d = tid & 7;
        float g = 0.0f;
        #pragma unroll
        for (int m = 0; m < 4; ++m) g += fc_w[o * 4 + m] * W_out[m * 8 + d];
        s_G[tid] = g;
        s_carry[tid] = 0.0f;                // causal conv left padding = 0
    }

    // ---- per-thread scan constants: tid = sn + 16*sd -------------------
    const int   sd   = tid >> 4;            // d_inner index 0..7
    const int   sn   = tid & 15;            // d_state index 0..15
    const float A_dn = -__expf(A_log[sd * 16 + sn]);
    const float wdt  = W_dt[sd];
    const float bdt  = b_dt[sd];
    const float Dd   = Dp[sd];
    float h       = 0.0f;                   // recurrent state (VGPR resident)
    float t_carry = 0.0f;                   // prev chunk's last time (tid 0)

    wait_async_x();
    __syncthreads();

    for (int chunk = 0; chunk < NCHUNK; ++chunk) {
        const int   base = chunk * CHUNK;
        const int   cur  = chunk & 1;
        const float* sxc = s_x[cur];

        // ---- 0. prefetch next chunk's x into the other buffer ---------
        if (chunk + 1 < NCHUNK)
            load_x_chunk(xb + (size_t)(base + CHUNK) * 4, s_x[cur ^ 1], tid);

        // ---- 1. xz = x @ W_in^T  via V_WMMA_F32_16X16X4_F32 -----------
        {
            v2f a, w;
            // A (16x4): lanes<16 hold K=0,1; lanes>=16 hold K=2,3
            a.x = sxc[(tt0 + lidx) * XS + 2 * half];
            a.y = sxc[(tt0 + lidx) * XS + 2 * half + 1];
            // B (4x16): B[k][e] = W_in[e][k]
            w.x = s_Win[lidx * 4 + 2 * half];
            w.y = s_Win[lidx * 4 + 2 * half + 1];
            v8f c = {};
            c = __builtin_amdgcn_wmma_f32_16x16x4_f32(
                    false, a, false, w, (short)0, c, false, false);
            #pragma unroll
            for (int r = 0; r < 8; ++r)
                s_xz[(tt0 + r + 8 * half) * XZS + lidx] = c[r];
        }
        __syncthreads();

        // ---- 2. depthwise causal conv (D_CONV=4) + SiLU -> u ----------
        for (int i = tid; i < CHUNK * 8; i += NTHREADS) {
            int t = i >> 3, d = i & 7;
            float v = s_cb[d];
            #pragma unroll
            for (int j = 0; j < 4; ++j) {
                int ts = t - 3 + j;
                float xij = (ts >= 0) ? s_xz[ts * XZS + d]
                                      : s_carry[(ts + 3) * 8 + d];
                v += s_cw[d * 4 + j] * xij;
            }
            s_u[t * US + d] = v * fast_sigmoid(v);      // silu
        }
        __syncthreads();

        // ---- 3. conv carry from last 3 tokens (for next chunk) --------
        if (tid < 24) {
            int c = tid >> 3, d = tid & 7;
            s_carry[c * 8 + d] = s_xz[(CHUNK - 3 + c) * XZS + d];
        }

        // ---- 4. x_dbl = u @ W_xproj^T (K=8 -> 2 steps, N=48 -> 3 tiles)
        #pragma unroll
        for (int j = 0; j < 3; ++j) {
            v8f c = {};
            #pragma unroll
            for (int s = 0; s < 2; ++s) {
                v2f a, w;
                a.x = s_u[(tt0 + lidx) * US + 4 * s + 2 * half];
                a.y = s_u[(tt0 + lidx) * US + 4 * s + 2 * half + 1];
                w.x = s_Wxp[(16 * j + lidx) * 8 + 4 * s + 2 * half];
                w.y = s_Wxp[(16 * j + lidx) * 8 + 4 * s + 2 * half + 1];
                c = __builtin_amdgcn_wmma_f32_16x16x4_f32(
                        false, a, false, w, (short)0, c, false, false);
            }
            #pragma unroll
            for (int r = 0; r < 8; ++r)
                s_xd[(tt0 + r + 8 * half) * XDS + 16 * j + lidx] = c[r];
        }
        __syncthreads();

        // ---- 5. selective scan over the 64 tokens ---------------------
        // thread (sd,sn) owns h[sd][sn]; y_d = sum_n h*C via 16-lane bfly
        for (int t = 0; t < CHUNK; ++t) {
            float dtr = s_xd[t * XDS + 0];              // dt_rank = 1
            float dtv = fast_softplus(dtr * wdt + bdt);
            float Bv  = s_xd[t * XDS + 1 + sn];
            float Cv  = s_xd[t * XDS + 17 + sn];
            float uv  = s_u[t * US + sd];

            h = __expf(dtv * A_dn) * h + (dtv * uv) * Bv;

            float p = h * Cv;
            p += __shfl_xor(p, 1, 32);
            p += __shfl_xor(p, 2, 32);
            p += __shfl_xor(p, 4, 32);
            p += __shfl_xor(p, 8, 32);                  // sum over d_state
            if (sn == 0) {
                float y  = p + uv * Dd;
                float zv = s_xz[t * XZS + 8 + sd];
                y *= zv * fast_sigmoid(zv);             // y * silu(z)
                s_y[t * 8 + sd] = y;
            }
        }
        __syncthreads();

        // ---- 6. epilogue: vel = y@G^T + fc_b; out = x[1:4] + vel*dtm --
        if (tid < CHUNK) {
            int   t  = tid;
            float tm = sxc[t * XS];
            float tp = (t > 0) ? sxc[(t - 1) * XS] : t_carry;
            float dtm = (base + t == 0) ? 0.0f : (tm - tp);
            float* op = out + ((size_t)b * SEQ_L + base + t) * 3;
            #pragma unroll
            for (int o = 0; o < 3; ++o) {
                float vel = fc_b[o];
                #pragma unroll
                for (int d = 0; d < 8; ++d)
                    vel += s_y[t * 8 + d] * s_G[o * 8 + d];
                op[o] = sxc[t * XS + 1 + o] + vel * dtm;
            }
            if (t == 0) t_carry = sxc[(CHUNK - 1) * XS]; // only tid 0 uses it
        }

        wait_async_x();          // next chunk's x tile landed in LDS
        __syncthreads();
    }
}

extern "C" void kernel_launch(void* const* d_in, const int* in_sizes, int n_in,
                              void* d_out, int out_size, void* d_ws, size_t ws_size,
                              hipStream_t stream) {
    (void)in_sizes; (void)n_in; (void)d_ws; (void)ws_size; (void)out_size;
    const float* x       = (const float*)d_in[0];
    const float* W_in    = (const float*)d_in[1];
    const float* conv_w  = (const float*)d_in[2];
    const float* conv_b  = (const float*)d_in[3];
    const float* W_xproj = (const float*)d_in[4];
    const float* W_dt    = (const float*)d_in[5];
    const float* b_dt    = (const float*)d_in[6];
    const float* A_log   = (const float*)d_in[7];
    const float* Dp      = (const float*)d_in[8];
    const float* W_out   = (const float*)d_in[9];
    const float* fc_w    = (const float*)d_in[10];
    const float* fc_b    = (const float*)d_in[11];
    float* out = (float*)d_out;

    mamba_fused_kernel<<<B_SZ, NTHREADS, 0, stream>>>(
        x, W_in, conv_w, conv_b, W_xproj, W_dt, b_dt, A_log, Dp,
        W_out, fc_w, fc_b, out);
}